// GMM_20650202759489
// MI455X (gfx1250) — compile-verified
//
#include <hip/hip_runtime.h>
#include <hip/hip_bf16.h>

typedef __attribute__((ext_vector_type(2))) float v2f;
typedef __attribute__((ext_vector_type(8))) float v8f;

#define N_PTS 100000
#define C_CL  16
#define D_DIM 64
#define TILES_PER_WG 8
#define ROW_STRIDE 68                 // 64 + 4 pad -> conflict-free ds_load_b64 fragments
#define TILE_STRIDE (16 * ROW_STRIDE) // 1088 floats per 16-row tile

// ds_swizzle_b32 xor-butterfly, pattern as instruction immediate (no index VALU).
// group-of-32 mode: imm = (xor_mask << 10) | (or_mask << 5) | and_mask
template <int IMM>
__device__ __forceinline__ float ds_swz(float x)
{
    return __builtin_bit_cast(
        float, __builtin_amdgcn_ds_swizzle(__builtin_bit_cast(int, x), IMM));
}

// max/sum butterflies over each 16-lane half (xor 1,2,4,8 stay within halves)
__device__ __forceinline__ float half16_max(float v)
{
    v = fmaxf(v, ds_swz<0x041F>(v));   // xor 1
    v = fmaxf(v, ds_swz<0x081F>(v));   // xor 2
    v = fmaxf(v, ds_swz<0x101F>(v));   // xor 4
    v = fmaxf(v, ds_swz<0x201F>(v));   // xor 8
    return v;
}
__device__ __forceinline__ float half16_sum(float v)
{
    v += ds_swz<0x041F>(v);
    v += ds_swz<0x081F>(v);
    v += ds_swz<0x101F>(v);
    v += ds_swz<0x201F>(v);
    return v;
}

__global__ __launch_bounds__(256)
void gmm_main(const float* __restrict__ data,
              const float* __restrict__ u,
              const float* __restrict__ log_var,
              const float* __restrict__ log_class_prior,
              const float* __restrict__ log_q_z_x,
              float* __restrict__ partials,
              int numTiles)
{
    __shared__ __align__(16) float sData[TILES_PER_WG * TILE_STRIDE]; // 34 KB
    __shared__ __align__(16) float sU[TILE_STRIDE];                   // 4.25 KB
    __shared__ float sDnp[256];     // per-thread |x|^2 partials (2 per row)
    __shared__ float sE[C_CL];
    __shared__ float sInv2Var[C_CL];
    __shared__ float sCterm[C_CL];
    __shared__ float sUnorm[C_CL];
    __shared__ float sWave[TILES_PER_WG];

    const int tid  = threadIdx.x;
    const int lane = tid & 31;
    const int wave = tid >> 5;
    const int half = lane >> 4;   // 0: lanes 0-15, 1: lanes 16-31
    const int sub  = lane & 15;

    // ---- Stage 8 tiles (128 rows x 64 f32 = 32 KB, contiguous) with b128 loads;
    //      accumulate squared-norm partials (each thread covers half a row). ----
    {
        const long blockStart = (long)blockIdx.x * (TILES_PER_WG * 16 * D_DIM);
        const int rowInBlk = tid >> 1;          // 0..127
        const int colBase  = (tid & 1) * 32;
        const int t8  = rowInBlk >> 4;
        const int rit = rowInBlk & 15;
        float* dst = &sData[t8 * TILE_STRIDE + rit * ROW_STRIDE + colBase];
        const long g0 = blockStart + (long)rowInBlk * D_DIM + colBase;
        float nsum = 0.0f;
#pragma unroll
        for (int j = 0; j < 8; ++j) {
            const long g = g0 + j * 4;
            float4 v = (g + 3 < (long)N_PTS * D_DIM)
                         ? *reinterpret_cast<const float4*>(data + g)
                         : make_float4(0.f, 0.f, 0.f, 0.f);
            dst[j*4+0] = v.x; dst[j*4+1] = v.y; dst[j*4+2] = v.z; dst[j*4+3] = v.w;
            nsum += v.x*v.x + v.y*v.y + v.z*v.z + v.w*v.w;
        }
        sDnp[tid] = nsum;
    }

    // ---- Stage u (16x64) into padded LDS ----
    if (tid < 64) {
        const int row = tid >> 2;
        const int colBase = (tid & 3) * 16;
#pragma unroll
        for (int j = 0; j < 4; ++j) {
            float4 v = *reinterpret_cast<const float4*>(u + row * D_DIM + colBase + j * 4);
            float* d = &sU[row * ROW_STRIDE + colBase + j * 4];
            d[0] = v.x; d[1] = v.y; d[2] = v.z; d[3] = v.w;
        }
    }

    // ---- Per-class precompute, part 1 (accurate exp here: only 16 lanes, once) ----
    if (tid < C_CL) {
        const float lv = log_var[tid];
        sInv2Var[tid] = 0.5f * expf(-lv);          // 1 / (2 e^{lv})
        sE[tid] = expf(log_class_prior[tid]);      // exp(log_class_prior)
        float un = 0.0f;
#pragma unroll
        for (int j = 0; j < 16; ++j) {
            float4 v = *reinterpret_cast<const float4*>(u + tid * D_DIM + j * 4);
            un += v.x*v.x + v.y*v.y + v.z*v.z + v.w*v.w;
        }
        sUnorm[tid] = un;                          // |u_c|^2
    }
    __syncthreads();

    // ---- Per-class precompute, part 2: lcp = log softmax(sE); cterm = lcp - 32*lv ----
    if (tid < C_CL) {
        float m = sE[0];
#pragma unroll
        for (int j = 1; j < C_CL; ++j) m = fmaxf(m, sE[j]);
        float s = 0.0f;
#pragma unroll
        for (int j = 0; j < C_CL; ++j) s += expf(sE[j] - m);
        const float lcp = sE[tid] - m - logf(s);
        sCterm[tid] = lcp - log_var[tid] * (0.5f * (float)D_DIM);
    }
    __syncthreads();

    // ---- Per-wave tile: WMMA f32 16x16x4 cross-term + fused epilogue ----
    const int tile = blockIdx.x * TILES_PER_WG + wave;   // wave-uniform
    float lsum = 0.0f;
    if (tile < numTiles) {
        const float* tb = &sData[wave * TILE_STRIDE];
        v8f acc = {0.f, 0.f, 0.f, 0.f, 0.f, 0.f, 0.f, 0.f};
        const int kb = half * 2;     // lanes 0-15: K=k0,k0+1 ; lanes 16-31: K=k0+2,k0+3
#pragma unroll
        for (int kk = 0; kk < 16; ++kk) {
            const int k0 = kk * 4 + kb;
            v2f a = *reinterpret_cast<const v2f*>(tb + sub * ROW_STRIDE + k0); // A = data tile
            v2f b = *reinterpret_cast<const v2f*>(&sU[sub * ROW_STRIDE + k0]); // B = u^T
            acc = __builtin_amdgcn_wmma_f32_16x16x4_f32(
                      false, a, false, b, (short)0, acc, false, false);
        }
        // acc[r] = dot(data[n0 + r + 8*half], u[sub])
        const float inv2v = sInv2Var[sub];
        const float ct    = sCterm[sub];
        const float un    = sUnorm[sub];
#pragma unroll
        for (int r = 0; r < 8; ++r) {
            const int M  = r + half * 8;
            const int br = wave * 16 + M;
            const float dn = sDnp[2 * br] + sDnp[2 * br + 1];      // |x_n|^2
            const float sq = fmaf(-2.0f, acc[r], dn + un);         // |x - u|^2
            const float y  = fmaf(-sq, inv2v, ct);                 // log_p + lcp
            const long  n  = (long)tile * 16 + M;
            // row softmax of exp(log_q_z_x[n,:]) across 16 lanes of this half-wave.
            // Native v_exp_f32 (TRANS32, co-executes with VALU); ds_swizzle
            // immediate-pattern butterflies (no index setup VALU).
            const float f  = __expf(log_q_z_x[n * C_CL + sub]);
            const float mx = half16_max(f);
            const float ef = __expf(f - mx);
            const float s  = half16_sum(ef);
            const float w  = __expf(ef / s);                       // exp(q_z_x)
            lsum = fmaf(y, w, lsum);
        }
    }

    // ---- deterministic block reduction (full-wave xor butterflies) ----
    lsum += ds_swz<0x041F>(lsum);
    lsum += ds_swz<0x081F>(lsum);
    lsum += ds_swz<0x101F>(lsum);
    lsum += ds_swz<0x201F>(lsum);
    lsum += ds_swz<0x401F>(lsum);      // xor 16: combine wave halves
    if (lane == 0) sWave[wave] = lsum;
    __syncthreads();
    if (tid == 0) {
        float t = 0.0f;
#pragma unroll
        for (int j = 0; j < TILES_PER_WG; ++j) t += sWave[j];
        partials[blockIdx.x] = t;
    }
}

__global__ __launch_bounds__(256)
void gmm_reduce(const float* __restrict__ ws, int n, float* __restrict__ out)
{
    __shared__ float s[256];
    float v = 0.0f;
    for (int i = threadIdx.x; i < n; i += 256) v += ws[i];
    s[threadIdx.x] = v;
    __syncthreads();
    for (int k = 128; k > 0; k >>= 1) {
        if (threadIdx.x < k) s[threadIdx.x] += s[threadIdx.x + k];
        __syncthreads();
    }
    if (threadIdx.x == 0) out[0] = -s[0];
}

extern "C" void kernel_launch(void* const* d_in, const int* in_sizes, int n_in,
                              void* d_out, int out_size, void* d_ws, size_t ws_size,
                              hipStream_t stream)
{
    const float* data = (const float*)d_in[0];
    const float* u    = (const float*)d_in[1];
    const float* lv   = (const float*)d_in[2];
    const float* lcp  = (const float*)d_in[3];
    const float* lq   = (const float*)d_in[4];
    float* out = (float*)d_out;
    float* ws  = (float*)d_ws;

    const int numTiles = (N_PTS + 15) / 16;                            // 6250
    const int numWG    = (numTiles + TILES_PER_WG - 1) / TILES_PER_WG; // 782

    gmm_main<<<numWG, 256, 0, stream>>>(data, u, lv, lcp, lq, ws, numTiles);
    gmm_reduce<<<1, 256, 0, stream>>>(ws, numWG, out);
}